// TextCNN_13469017440401
// MI455X (gfx1250) — compile-verified
//
#include <hip/hip_runtime.h>

typedef float v2f __attribute__((ext_vector_type(2)));
typedef float v8f __attribute__((ext_vector_type(8)));

#define L_SEQ   4096
#define E_DIM   128
#define FN_     128
#define B_BATCH 32
#define TILE_P  64
#define HALO    4
#define ROWS    (TILE_P + HALO)     // 68
#define LDS_ST  132                 // padded row stride (floats) -> conflict-free banks
#define NS      3

// ---------------------------------------------------------------------------
// Per-filter-size WMMA GEMM + fused attention epilogue.
// A (16x4)  : positions x K, from LDS (im2col = address shift by i)
// B (4x16)  : K x filters, from global (weights are L2-resident)
// C (16x16) : positions x filters, fp32 accumulators
// ---------------------------------------------------------------------------
template <int LF>
__device__ __forceinline__ float conv_partial(const float* __restrict__ wc,
                                              const float* __restrict__ bc,
                                              const float* __restrict__ wa,
                                              const float* __restrict__ ba,
                                              const float* wt,   // wterm[64] (LDS)
                                              const float* wl,   // masked wl[64] (LDS)
                                              const float* a_base, // LDS base for this lane
                                              int half, int fcol)
{
    v8f acc[4] = {};
    // B fragment source: wc[f, i, e], f = fcol ; per lane: e4 + 2*half + {0,1}
    const float* wb = wc + (size_t)fcol * (LF * E_DIM) + 2 * half;

    #pragma unroll 8
    for (int kk = 0; kk < 32 * LF; ++kk) {
        const int i  = kk >> 5;          // which halo row of the window
        const int e4 = (kk & 31) << 2;   // embedding offset (steps of 4)
        v2f bf = *(const v2f*)(wb + i * E_DIM + e4);
        const float* ar = a_base + i * LDS_ST + e4;
        #pragma unroll
        for (int m = 0; m < 4; ++m) {
            v2f af = *(const v2f*)(ar + m * 16 * LDS_ST);
            acc[m] = __builtin_amdgcn_wmma_f32_16x16x4_f32(
                false, af, false, bf, (short)0, acc[m], false, false);
        }
    }

    // Epilogue: xv=relu(conv+bc); att=sigmoid(xv*wa0 + wterm + ba); h=relu(xv*att);
    // partial = sum_p h * wl[p]   (wl pre-masked to 0 for p >= P_l)
    const float bcf  = bc[fcol];
    const float wa0s = wa[0];
    const float bas  = ba[0];
    float partial = 0.f;
    #pragma unroll
    for (int m = 0; m < 4; ++m) {
        #pragma unroll
        for (int j = 0; j < 8; ++j) {
            const int pl = m * 16 + j + half * 8;     // C/D VGPR layout
            float xv  = fmaxf(acc[m][j] + bcf, 0.f);
            float z   = fmaf(xv, wa0s, wt[pl] + bas);
            float att = 1.f / (1.f + __expf(-z));
            partial   = fmaf(fmaxf(xv * att, 0.f), wl[pl], partial);
        }
    }
    return partial;
}

__launch_bounds__(256)
__global__ void textcnn_conv_kernel(
    const float* __restrict__ x,   // (B, L, E)
    const float* __restrict__ w1,  // (3, E)
    const float* __restrict__ b1,  // (3,)
    const float* __restrict__ wc0, const float* __restrict__ bc0,
    const float* __restrict__ wa0, const float* __restrict__ ba0,
    const float* __restrict__ wl0,
    const float* __restrict__ wc1, const float* __restrict__ bc1,
    const float* __restrict__ wa1, const float* __restrict__ ba1,
    const float* __restrict__ wl1,
    const float* __restrict__ wc2, const float* __restrict__ bc2,
    const float* __restrict__ wa2, const float* __restrict__ ba2,
    const float* __restrict__ wl2,
    float* __restrict__ S)         // (B, 3*FN) pooled accumulators
{
    __shared__ float xs[ROWS * LDS_ST];
    __shared__ float c1s[ROWS * 3];
    __shared__ float wterm[NS][TILE_P];
    __shared__ float wlv[NS][TILE_P];

    const int b    = blockIdx.y;
    const int p0   = blockIdx.x * TILE_P;
    const int tid  = threadIdx.x;
    const int lane = tid & 31;
    const int wave = tid >> 5;
    const int half = lane >> 4;
    const int l15  = lane & 15;
    const int fcol = wave * 16 + l15;   // this lane's filter column

    // ---- stage x tile (ROWS x 128) into LDS, rows clamped at L-1 ----
    const float* xb = x + (size_t)b * L_SEQ * E_DIM;
    for (int idx = tid; idx < ROWS * (E_DIM / 4); idx += 256) {
        int row  = idx >> 5;
        int c4   = (idx & 31) << 2;
        int grow = p0 + row; if (grow > L_SEQ - 1) grow = L_SEQ - 1;
        const float4 v = *(const float4*)(xb + (size_t)grow * E_DIM + c4);
        float* dst = &xs[row * LDS_ST + c4];
        dst[0] = v.x; dst[1] = v.y; dst[2] = v.z; dst[3] = v.w;
    }
    __syncthreads();

    // ---- c1 tile: c1[t,c] = dot(x[t,:], w1[c,:]) + b1[c] ----
    if (tid < ROWS * 3) {
        int row = tid / 3, c = tid - row * 3;
        const float* wr = w1 + c * E_DIM;
        const float* xr = &xs[row * LDS_ST];
        float acc = b1[c];
        #pragma unroll 8
        for (int e = 0; e < E_DIM; ++e) acc = fmaf(xr[e], wr[e], acc);
        c1s[tid] = acc;
    }
    __syncthreads();

    // ---- per-size window term + masked pooling weight ----
    if (tid < NS * TILE_P) {
        int s = tid >> 6;
        int p = tid & 63;
        int l = 3 + s;
        const float* wa = (s == 0) ? wa0 : (s == 1) ? wa1 : wa2;
        const float* wl = (s == 0) ? wl0 : (s == 1) ? wl1 : wl2;
        float acc = 0.f;
        for (int c = 0; c < 3; ++c)
            for (int i = 0; i < l; ++i)
                acc = fmaf(c1s[(p + i) * 3 + c], wa[1 + c * l + i], acc);
        wterm[s][p] = acc;
        int gp = p0 + p;
        int P  = L_SEQ - l + 1;
        wlv[s][p] = (gp < P) ? wl[gp] : 0.f;
    }
    __syncthreads();

    const float* a_base = &xs[l15 * LDS_ST + 2 * half];

    float r0 = conv_partial<3>(wc0, bc0, wa0, ba0, wterm[0], wlv[0], a_base, half, fcol);
    float r1 = conv_partial<4>(wc1, bc1, wa1, ba1, wterm[1], wlv[1], a_base, half, fcol);
    float r2 = conv_partial<5>(wc2, bc2, wa2, ba2, wterm[2], wlv[2], a_base, half, fcol);

    float* Sb = S + (size_t)b * (NS * FN_);
    atomicAdd(&Sb[0 * FN_ + fcol], r0);
    atomicAdd(&Sb[1 * FN_ + fcol], r1);
    atomicAdd(&Sb[2 * FN_ + fcol], r2);
}

// ---------------------------------------------------------------------------
// Head: feat = relu(S + bl_size); out[b] = sigmoid(feat @ wfc + bfc)
// ---------------------------------------------------------------------------
__global__ void textcnn_head_kernel(const float* __restrict__ S,
                                    const float* __restrict__ bl0,
                                    const float* __restrict__ bl1,
                                    const float* __restrict__ bl2,
                                    const float* __restrict__ wfc,
                                    const float* __restrict__ bfc,
                                    float* __restrict__ out)
{
    __shared__ float red[NS * FN_];
    const int b = blockIdx.x;
    const int t = threadIdx.x;   // 384 threads
    float bl = (t < FN_) ? bl0[0] : (t < 2 * FN_) ? bl1[0] : bl2[0];
    float feat = fmaxf(S[(size_t)b * (NS * FN_) + t] + bl, 0.f);
    red[t] = feat * wfc[t];
    __syncthreads();
    if (t < 128) red[t] += red[t + 128] + red[t + 256];
    __syncthreads();
    for (int off = 64; off > 0; off >>= 1) {
        if (t < off) red[t] += red[t + off];
        __syncthreads();
    }
    if (t == 0) {
        float z = red[0] + bfc[0];
        out[b] = 1.f / (1.f + __expf(-z));
    }
}

__global__ void zero_kernel(float* __restrict__ p, int n) {
    int i = blockIdx.x * blockDim.x + threadIdx.x;
    if (i < n) p[i] = 0.f;
}

extern "C" void kernel_launch(void* const* d_in, const int* in_sizes, int n_in,
                              void* d_out, int out_size, void* d_ws, size_t ws_size,
                              hipStream_t stream) {
    const float* x   = (const float*)d_in[0];
    const float* w1  = (const float*)d_in[1];
    const float* b1  = (const float*)d_in[2];
    const float* wc0 = (const float*)d_in[3];
    const float* bc0 = (const float*)d_in[4];
    const float* wa0 = (const float*)d_in[5];
    const float* ba0 = (const float*)d_in[6];
    const float* wl0 = (const float*)d_in[7];
    const float* bl0 = (const float*)d_in[8];
    const float* wc1 = (const float*)d_in[9];
    const float* bc1 = (const float*)d_in[10];
    const float* wa1 = (const float*)d_in[11];
    const float* ba1 = (const float*)d_in[12];
    const float* wl1 = (const float*)d_in[13];
    const float* bl1 = (const float*)d_in[14];
    const float* wc2 = (const float*)d_in[15];
    const float* bc2 = (const float*)d_in[16];
    const float* wa2 = (const float*)d_in[17];
    const float* ba2 = (const float*)d_in[18];
    const float* wl2 = (const float*)d_in[19];
    const float* bl2 = (const float*)d_in[20];
    const float* wfc = (const float*)d_in[21];
    const float* bfc = (const float*)d_in[22];

    float* S = (float*)d_ws;                      // B x (3*FN) fp32 accumulators
    const int nS = B_BATCH * NS * FN_;

    zero_kernel<<<(nS + 255) / 256, 256, 0, stream>>>(S, nS);

    dim3 grid(L_SEQ / TILE_P, B_BATCH);           // 64 x 32 = 2048 workgroups
    textcnn_conv_kernel<<<grid, 256, 0, stream>>>(
        x, w1, b1,
        wc0, bc0, wa0, ba0, wl0,
        wc1, bc1, wa1, ba1, wl1,
        wc2, bc2, wa2, ba2, wl2,
        S);

    textcnn_head_kernel<<<B_BATCH, NS * FN_, 0, stream>>>(
        S, bl0, bl1, bl2, wfc, bfc, (float*)d_out);
}